// Attention_46600395162125
// MI455X (gfx1250) — compile-verified
//
#include <hip/hip_runtime.h>
#include <math.h>

typedef __attribute__((ext_vector_type(2))) float v2f;
typedef __attribute__((ext_vector_type(4))) float v4f;
typedef __attribute__((ext_vector_type(8))) float v8f;
typedef __attribute__((ext_vector_type(4))) int   v4i;

#define HD    1024
#define BATCH 32
#define TLEN  1024
#define KC    128      // K-chunk staged in LDS
#define KPAD  132      // +4 floats pad: bank-conflict-free, 16B-aligned rows
#define NCH   16       // T-chunks for the context partial pass
#define TCH   (TLEN / NCH)

// gfx1250 async global->LDS path, if this toolchain exposes the builtins.
#if defined(__has_builtin)
#if __has_builtin(__builtin_amdgcn_global_load_async_to_lds_b128) && \
    __has_builtin(__builtin_amdgcn_s_wait_asynccnt)
#define USE_ASYNC_LDS 1
#endif
#endif

#ifdef USE_ASYNC_LDS
typedef __attribute__((address_space(1))) v4i gv4i;
typedef __attribute__((address_space(3))) v4i lv4i;
#endif

__device__ __forceinline__ float selu_f(float x) {
    const float alpha = 1.6732632423543772f;
    const float scale = 1.0507009873554805f;
    return x > 0.0f ? scale * x : scale * alpha * expm1f(x);
}

// Y(32,1024) = selu( X(32,1024) @ W(1024,1024)^T + bias )
// grid.x = 64 (16-col N tiles), block = 64 threads = 2 waves (one per 16-row M tile).
__global__ __launch_bounds__(64) void mlp_selu_wmma(
    const float* __restrict__ X,
    const float* __restrict__ W,
    const float* __restrict__ bias,
    float* __restrict__ Y)
{
    __shared__ float sA[BATCH * KPAD];  // 32 rows of X, one K chunk
    __shared__ float sW[16 * KPAD];     // 16 rows of W (this N tile), one K chunk

    const int tid  = threadIdx.x;
    const int lane = tid & 31;
    const int wv   = tid >> 5;          // 0..1 -> M rows [wv*16, wv*16+16)
    const int hi   = lane >> 4;         // 0: K pair 0..1, 1: K pair 2..3
    const int ln   = lane & 15;
    const int n0   = blockIdx.x * 16;

    v8f c = {};

    for (int kb = 0; kb < HD; kb += KC) {
        // stage X chunk: 32 x 128 floats = 1024 x 16B, coalesced
        for (int i = tid; i < BATCH * (KC / 4); i += 64) {
            int r  = i >> 5;            // 32 float4 per row
            int c4 = i & 31;
#ifdef USE_ASYNC_LDS
            __builtin_amdgcn_global_load_async_to_lds_b128(
                (gv4i*)(X + r * HD + kb + c4 * 4),
                (lv4i*)(sA + r * KPAD + c4 * 4), 0, 0);
#else
            v4f v = *(const v4f*)&X[r * HD + kb + c4 * 4];
            *(v4f*)&sA[r * KPAD + c4 * 4] = v;
#endif
        }
        // stage W chunk: 16 x 128 floats = 512 x 16B, coalesced
        for (int i = tid; i < 16 * (KC / 4); i += 64) {
            int r  = i >> 5;
            int c4 = i & 31;
#ifdef USE_ASYNC_LDS
            __builtin_amdgcn_global_load_async_to_lds_b128(
                (gv4i*)(W + (n0 + r) * HD + kb + c4 * 4),
                (lv4i*)(sW + r * KPAD + c4 * 4), 0, 0);
#else
            v4f v = *(const v4f*)&W[(n0 + r) * HD + kb + c4 * 4];
            *(v4f*)&sW[r * KPAD + c4 * 4] = v;
#endif
        }
#ifdef USE_ASYNC_LDS
        __builtin_amdgcn_s_wait_asynccnt(0);
#endif
        __syncthreads();

        const float* aRow = &sA[(wv * 16 + ln) * KPAD];
        const float* bRow = &sW[ln * KPAD];
        #pragma unroll 8
        for (int k0 = 0; k0 < KC; k0 += 4) {
            v2f a = *(const v2f*)&aRow[k0 + 2 * hi];
            v2f b = *(const v2f*)&bRow[k0 + 2 * hi];
            c = __builtin_amdgcn_wmma_f32_16x16x4_f32(
                    false, a, false, b, (short)0, c, false, false);
        }
        __syncthreads();
    }

    // D layout: c[v] -> row M = v + 8*hi within this wave's 16-row tile, col = ln
    const float bn = bias[n0 + ln];
    #pragma unroll
    for (int v = 0; v < 8; ++v) {
        int m = wv * 16 + v + 8 * hi;
        Y[m * HD + n0 + ln] = selu_f(c[v] + bn);
    }
}

// scores[b,t] = dot(E[b,t,:], h[b,:]); one wave per (b,t) row
__global__ __launch_bounds__(256) void scores_dot(
    const float* __restrict__ E,    // (32,1024,1024)
    const float* __restrict__ h,    // (32,1024)
    float* __restrict__ S)          // (32,1024)
{
    const int lane = threadIdx.x & 31;
    const int gw   = blockIdx.x * 8 + (threadIdx.x >> 5);
    const int b    = gw >> 10;
    const int t    = gw & 1023;
    const float* e  = &E[(size_t)b * TLEN * HD + (size_t)t * HD];
    const float* hb = &h[b * HD];
    float acc = 0.0f;
    #pragma unroll
    for (int j = 0; j < 8; ++j) {
        int k = j * 128 + lane * 4;
        v4f ev = *(const v4f*)&e[k];
        v4f hv = *(const v4f*)&hb[k];
        acc = fmaf(ev.x, hv.x, acc);
        acc = fmaf(ev.y, hv.y, acc);
        acc = fmaf(ev.z, hv.z, acc);
        acc = fmaf(ev.w, hv.w, acc);
    }
    for (int off = 16; off > 0; off >>= 1) acc += __shfl_xor(acc, off, 32);
    if (lane == 0) S[b * TLEN + t] = acc;
}

// softmax over T per batch; writes normalized weights
__global__ __launch_bounds__(1024) void softmax_t(
    const float* __restrict__ S,    // (32,1024)
    float* __restrict__ Wt)         // (32,1024)
{
    __shared__ float red[32];
    const int b    = blockIdx.x;
    const int t    = threadIdx.x;
    const int lane = t & 31;
    const int wv   = t >> 5;

    float s = S[b * TLEN + t];

    // block max
    float m = s;
    for (int off = 16; off > 0; off >>= 1) m = fmaxf(m, __shfl_xor(m, off, 32));
    if (lane == 0) red[wv] = m;
    __syncthreads();
    if (wv == 0) {
        float v = red[lane];
        for (int off = 16; off > 0; off >>= 1) v = fmaxf(v, __shfl_xor(v, off, 32));
        if (lane == 0) red[0] = v;
    }
    __syncthreads();
    m = red[0];
    __syncthreads();

    // block sum of exp
    float e = expf(s - m);
    float sum = e;
    for (int off = 16; off > 0; off >>= 1) sum += __shfl_xor(sum, off, 32);
    if (lane == 0) red[wv] = sum;
    __syncthreads();
    if (wv == 0) {
        float v = red[lane];
        for (int off = 16; off > 0; off >>= 1) v += __shfl_xor(v, off, 32);
        if (lane == 0) red[0] = v;
    }
    __syncthreads();
    sum = red[0];

    Wt[b * TLEN + t] = e / sum;
}

// Partial context over a 64-row T chunk: P[b,chunk,:] = sum_t w[t]*E[b,t,:]
// grid (16 chunks, 32 batches), block 256, 4 cols (float4) per thread.
__global__ __launch_bounds__(256) void context_partial(
    const float* __restrict__ E,
    const float* __restrict__ Wt,
    float* __restrict__ P)          // (32, 16, 1024)
{
    __shared__ float wb[TCH];
    const int tc = blockIdx.x;
    const int b  = blockIdx.y;
    const int t0 = tc * TCH;
    if (threadIdx.x < TCH) wb[threadIdx.x] = Wt[b * TLEN + t0 + threadIdx.x];
    __syncthreads();

    const float* eb = &E[(size_t)b * TLEN * HD + (size_t)t0 * HD];
    const int col = threadIdx.x * 4;
    v4f acc = {};
    #pragma unroll 4
    for (int t = 0; t < TCH; ++t) {
        v4f ev = *(const v4f*)&eb[(size_t)t * HD + col];
        float w = wb[t];
        acc.x = fmaf(w, ev.x, acc.x);
        acc.y = fmaf(w, ev.y, acc.y);
        acc.z = fmaf(w, ev.z, acc.z);
        acc.w = fmaf(w, ev.w, acc.w);
    }
    *(v4f*)&P[((size_t)b * NCH + tc) * HD + col] = acc;
}

// C[b,:] = sum_chunks P[b,chunk,:]; grid 32, block 256, float4 per thread
__global__ __launch_bounds__(256) void context_reduce(
    const float* __restrict__ P,
    float* __restrict__ C)
{
    const int b   = blockIdx.x;
    const int col = threadIdx.x * 4;
    v4f acc = {};
    #pragma unroll
    for (int c = 0; c < NCH; ++c) {
        v4f v = *(const v4f*)&P[((size_t)b * NCH + c) * HD + col];
        acc.x += v.x; acc.y += v.y; acc.z += v.z; acc.w += v.w;
    }
    *(v4f*)&C[(size_t)b * HD + col] = acc;
}

// Fallback (small workspace): one block per batch, full T loop, float4 cols.
__global__ __launch_bounds__(256) void context_direct(
    const float* __restrict__ E,
    const float* __restrict__ Wt,
    float* __restrict__ C)
{
    __shared__ float wb[TLEN];
    const int b   = blockIdx.x;
    const int col = threadIdx.x * 4;
    for (int i = threadIdx.x; i < TLEN; i += 256) wb[i] = Wt[b * TLEN + i];
    __syncthreads();

    const float* eb = &E[(size_t)b * TLEN * HD];
    v4f acc = {};
    #pragma unroll 4
    for (int t = 0; t < TLEN; ++t) {
        if ((t & 63) == 0)
            __builtin_prefetch(&eb[(size_t)(t + 64) * HD + col], 0, 0);
        v4f ev = *(const v4f*)&eb[(size_t)t * HD + col];
        float w = wb[t];
        acc.x = fmaf(w, ev.x, acc.x);
        acc.y = fmaf(w, ev.y, acc.y);
        acc.z = fmaf(w, ev.z, acc.z);
        acc.w = fmaf(w, ev.w, acc.w);
    }
    *(v4f*)&C[(size_t)b * HD + col] = acc;
}

extern "C" void kernel_launch(void* const* d_in, const int* in_sizes, int n_in,
                              void* d_out, int out_size, void* d_ws, size_t ws_size,
                              hipStream_t stream) {
    (void)in_sizes; (void)n_in; (void)out_size;
    const float* hidden = (const float*)d_in[0];   // (1,32,1024)
    const float* E      = (const float*)d_in[1];   // (32,1024,1024)
    const float* W1     = (const float*)d_in[2];   // (1024,1024)
    const float* b1     = (const float*)d_in[3];   // (1024)
    const float* W2     = (const float*)d_in[4];   // (1024,1024)
    const float* b2     = (const float*)d_in[5];   // (1024)

    float* out = (float*)d_out;                    // [weights(32*1024) | context(32*1024)]
    float* ws  = (float*)d_ws;
    float* h1  = ws;                               // (32,1024)
    float* h2  = ws + BATCH * HD;                  // (32,1024)
    float* sc  = ws + 2 * BATCH * HD;              // (32,1024)
    float* pc  = ws + 3 * BATCH * HD;              // (32,16,1024) partials

    const size_t need = (size_t)(3 * BATCH * HD + BATCH * NCH * HD) * sizeof(float);

    mlp_selu_wmma<<<dim3(HD / 16), dim3(64), 0, stream>>>(hidden, W1, b1, h1);
    mlp_selu_wmma<<<dim3(HD / 16), dim3(64), 0, stream>>>(h1, W2, b2, h2);

    scores_dot<<<dim3((BATCH * TLEN) / 8), dim3(256), 0, stream>>>(E, h2, sc);
    softmax_t<<<dim3(BATCH), dim3(1024), 0, stream>>>(sc, out);

    float* ctx = out + BATCH * TLEN;
    if (ws_size >= need) {
        context_partial<<<dim3(NCH, BATCH), dim3(256), 0, stream>>>(E, out, pc);
        context_reduce<<<dim3(BATCH), dim3(256), 0, stream>>>(pc, ctx);
    } else {
        context_direct<<<dim3(BATCH), dim3(256), 0, stream>>>(E, out, ctx);
    }
}